// HeteroGCN_dgl_17119739641941
// MI455X (gfx1250) — compile-verified
//
#include <hip/hip_runtime.h>
#include <math.h>

// ---------------------------------------------------------------------------
// HeteroGCN (DGL GraphConv norm='both', HeteroGraphConv mean-aggregate) for
// MI455X / gfx1250.  fp32 end-to-end; GEMMs run on V_WMMA_F32_16X16X4_F32.
// ---------------------------------------------------------------------------

typedef __attribute__((ext_vector_type(2))) float v2f;
typedef __attribute__((ext_vector_type(8))) float v8f;

#define NODES 50000
#define EDGES 200000
#define RELS  3
#define FEAT  128
#define HID   256
#define PAIRS 100000

// ---------------------------------------------------------------------------
__global__ void zero_f32(float* __restrict__ p, int n) {
  int i = blockIdx.x * blockDim.x + threadIdx.x;
  if (i < n) p[i] = 0.0f;
}

// deg_out[r][src]++ , deg_in[r][dst]++  over all relations.
__global__ void degree_kernel(const int* __restrict__ ei,
                              float* __restrict__ dout,
                              float* __restrict__ din) {
  int t = blockIdx.x * blockDim.x + threadIdx.x;
  if (t >= RELS * EDGES) return;
  int r = t / EDGES, e = t - r * EDGES;
  int s = ei[(size_t)r * 2 * EDGES + e];
  int d = ei[(size_t)r * 2 * EDGES + EDGES + e];
  atomicAdd(&dout[r * NODES + s], 1.0f);
  atomicAdd(&din[r * NODES + d], 1.0f);
}

// agg[dst] += X[src] * rsqrt(max(deg_out[src],1)); one thread per (edge, 4-feat)
__global__ void scatter_agg(const float* __restrict__ X,
                            const int* __restrict__ src,
                            const int* __restrict__ dst,
                            const float* __restrict__ dout,
                            float* __restrict__ agg, int Fdim) {
  int t = blockIdx.x * blockDim.x + threadIdx.x;
  int groups = Fdim >> 2;
  int e = t / groups;
  if (e >= EDGES) return;
  int g = t - e * groups;
  int s = src[e], d = dst[e];
  float sc = rsqrtf(fmaxf(dout[s], 1.0f));
  float4 xv = ((const float4*)(X + (size_t)s * Fdim))[g];
  float* o = agg + (size_t)d * Fdim + (g << 2);
  atomicAdd(o + 0, xv.x * sc);
  atomicAdd(o + 1, xv.y * sc);
  atomicAdd(o + 2, xv.z * sc);
  atomicAdd(o + 3, xv.w * sc);
}

// C[16x64 tile] += diag(rsqrt(max(deg_in,1))) * A[MxK] @ B[KxNcols]
// One wave per tile; fp32 WMMA 16x16x4.  A/B/C row-major.
// ISA 7.12.2 fragment layouts:
//   A 16x4 : lanes 0-15 -> M=lane, VGPR{0,1}=K{0,1}; lanes 16-31 -> K{2,3}
//   B 4x16 : lanes 0-15 -> N=lane, VGPR{0,1}=K{0,1}; lanes 16-31 -> K{2,3}
//   C/D    : VGPR v -> M = v (+8 for lanes 16-31), N = lane & 15
__global__ void gemm_wmma(const float* __restrict__ A,
                          const float* __restrict__ B,
                          const float* __restrict__ din,
                          float* __restrict__ C,
                          int M, int K, int Ncols) {
  int wave = (blockIdx.x * blockDim.x + threadIdx.x) >> 5;
  int lane = threadIdx.x & 31;
  int nTiles = Ncols >> 6;                 // 64 columns per wave
  int mTile = wave / nTiles;
  int nTile = wave - mTile * nTiles;
  if (mTile * 16 >= M) return;             // wave-uniform: EXEC all-ones below
  int row0 = mTile * 16, col0 = nTile * 64;

  int mA   = lane & 15;
  int koff = (lane >> 4) << 1;             // 0 for lanes 0-15, 2 for 16-31
  int rowA = row0 + mA;
  float s  = rsqrtf(fmaxf(din[rowA], 1.0f));
  const float* Ap = A + (size_t)rowA * K + koff;
  const float* Bp = B + col0 + (lane & 15);

  v8f acc0 = {0,0,0,0,0,0,0,0};
  v8f acc1 = acc0, acc2 = acc0, acc3 = acc0;

  for (int k0 = 0; k0 < K; k0 += 4) {
    v2f a;
    a.x = Ap[k0]     * s;
    a.y = Ap[k0 + 1] * s;
    const float* bp = Bp + (size_t)(k0 + koff) * Ncols;
    v2f b0, b1, b2, b3;
    b0.x = bp[0];  b0.y = bp[Ncols];
    b1.x = bp[16]; b1.y = bp[Ncols + 16];
    b2.x = bp[32]; b2.y = bp[Ncols + 32];
    b3.x = bp[48]; b3.y = bp[Ncols + 48];
    acc0 = __builtin_amdgcn_wmma_f32_16x16x4_f32(false, a, false, b0, (short)0, acc0, false, false);
    acc1 = __builtin_amdgcn_wmma_f32_16x16x4_f32(false, a, false, b1, (short)0, acc1, false, false);
    acc2 = __builtin_amdgcn_wmma_f32_16x16x4_f32(false, a, false, b2, (short)0, acc2, false, false);
    acc3 = __builtin_amdgcn_wmma_f32_16x16x4_f32(false, a, false, b3, (short)0, acc3, false, false);
  }

  int cRow = row0 + ((lane >> 4) << 3);    // +8 rows for upper half-wave
  int cCol = lane & 15;
  float* cp = C + (size_t)cRow * Ncols + col0 + cCol;
#pragma unroll
  for (int v = 0; v < 8; ++v) {
    cp[(size_t)v * Ncols + 0]  += acc0[v];
    cp[(size_t)v * Ncols + 16] += acc1[v];
    cp[(size_t)v * Ncols + 32] += acc2[v];
    cp[(size_t)v * Ncols + 48] += acc3[v];
  }
}

// acc = maybe_relu(acc/3 + mean_r bias[r]); in place.
__global__ void finalize(float* __restrict__ acc, const float* __restrict__ bias,
                         int D, int doRelu) {
  int i = blockIdx.x * blockDim.x + threadIdx.x;
  if (i >= NODES * D) return;
  int j = i % D;
  float mb = (bias[j] + bias[D + j] + bias[2 * D + j]) * (1.0f / 3.0f);
  float v = acc[i] * (1.0f / 3.0f) + mb;
  acc[i] = doRelu ? fmaxf(v, 0.0f) : v;
}

// One wave32 per output row: sigmoid(relu([h[s],h[d]]) . Wlin + blin)
__global__ void predict(const float* __restrict__ h, const int* __restrict__ ei,
                        const int* __restrict__ npairs,
                        const float* __restrict__ Wlin,
                        const float* __restrict__ blin,
                        float* __restrict__ out) {
  int wid = (blockIdx.x * blockDim.x + threadIdx.x) >> 5;
  int lane = threadIdx.x & 31;
  const int total = RELS * EDGES + PAIRS;
  if (wid >= total) return;
  int s, d;
  if (wid < RELS * EDGES) {
    int r = wid / EDGES, e = wid - r * EDGES;
    s = ei[(size_t)r * 2 * EDGES + e];
    d = ei[(size_t)r * 2 * EDGES + EDGES + e];
  } else {
    int p = wid - RELS * EDGES;
    s = npairs[2 * p];
    d = npairs[2 * p + 1];
  }
  const float* hs = h + (size_t)s * FEAT;
  const float* hd = h + (size_t)d * FEAT;
  float acc = 0.0f;
  for (int j = lane; j < FEAT; j += 32) {
    acc += fmaxf(hs[j], 0.0f) * Wlin[j];
    acc += fmaxf(hd[j], 0.0f) * Wlin[FEAT + j];
  }
  for (int off = 16; off > 0; off >>= 1) acc += __shfl_xor(acc, off, 32);
  if (lane == 0) out[wid] = 1.0f / (1.0f + expf(-(acc + blin[0])));
}

// ---------------------------------------------------------------------------
extern "C" void kernel_launch(void* const* d_in, const int* in_sizes, int n_in,
                              void* d_out, int out_size, void* d_ws, size_t ws_size,
                              hipStream_t stream) {
  (void)in_sizes; (void)n_in; (void)out_size; (void)ws_size;
  const float* x     = (const float*)d_in[0];
  const int*   ei    = (const int*)  d_in[1];
  const int*   npair = (const int*)  d_in[2];
  const float* W1    = (const float*)d_in[3];
  const float* b1    = (const float*)d_in[4];
  const float* W2    = (const float*)d_in[5];
  const float* b2    = (const float*)d_in[6];
  const float* Wlin  = (const float*)d_in[7];
  const float* blin  = (const float*)d_in[8];
  float* out = (float*)d_out;

  char* w = (char*)d_ws;
  float* deg_out = (float*)w; w += (size_t)RELS * NODES * 4;
  float* deg_in  = (float*)w; w += (size_t)RELS * NODES * 4;
  float* agg     = (float*)w; w += (size_t)NODES * HID * 4;   // reused both layers
  float* acc1    = (float*)w; w += (size_t)NODES * HID * 4;   // becomes h1
  float* acc2    = (float*)w; w += (size_t)NODES * FEAT * 4;  // becomes h2

  const int B = 256;
  auto nb = [](long long n, int b) { return (int)((n + b - 1) / b); };

  // degrees (deg_out/deg_in contiguous -> single zero pass)
  zero_f32<<<nb(2LL * RELS * NODES, B), B, 0, stream>>>(deg_out, 2 * RELS * NODES);
  degree_kernel<<<nb((long long)RELS * EDGES, B), B, 0, stream>>>(ei, deg_out, deg_in);

  // ----- layer 1: h1 = relu(mean_r GraphConv_r(x; W1_r, b1_r)) -----
  zero_f32<<<nb((long long)NODES * HID, B), B, 0, stream>>>(acc1, NODES * HID);
  for (int r = 0; r < RELS; ++r) {
    const int* src = ei + (size_t)r * 2 * EDGES;
    const int* dst = src + EDGES;
    zero_f32<<<nb((long long)NODES * FEAT, B), B, 0, stream>>>(agg, NODES * FEAT);
    scatter_agg<<<nb((long long)EDGES * (FEAT / 4), B), B, 0, stream>>>(
        x, src, dst, deg_out + r * NODES, agg, FEAT);
    long long waves = (long long)(NODES / 16) * (HID / 64);
    gemm_wmma<<<nb(waves * 32, B), B, 0, stream>>>(
        agg, W1 + (size_t)r * FEAT * HID, deg_in + r * NODES, acc1, NODES, FEAT, HID);
  }
  finalize<<<nb((long long)NODES * HID, B), B, 0, stream>>>(acc1, b1, HID, 1);

  // ----- layer 2: h2 = mean_r GraphConv_r(h1; W2_r, b2_r) -----
  zero_f32<<<nb((long long)NODES * FEAT, B), B, 0, stream>>>(acc2, NODES * FEAT);
  for (int r = 0; r < RELS; ++r) {
    const int* src = ei + (size_t)r * 2 * EDGES;
    const int* dst = src + EDGES;
    zero_f32<<<nb((long long)NODES * HID, B), B, 0, stream>>>(agg, NODES * HID);
    scatter_agg<<<nb((long long)EDGES * (HID / 4), B), B, 0, stream>>>(
        acc1, src, dst, deg_out + r * NODES, agg, HID);
    long long waves = (long long)(NODES / 16) * (FEAT / 64);
    gemm_wmma<<<nb(waves * 32, B), B, 0, stream>>>(
        agg, W2 + (size_t)r * HID * FEAT, deg_in + r * NODES, acc2, NODES, HID, FEAT);
  }
  finalize<<<nb((long long)NODES * FEAT, B), B, 0, stream>>>(acc2, b2, FEAT, 0);

  // ----- edge/pair scoring -----
  long long total = (long long)RELS * EDGES + PAIRS;
  predict<<<nb(total * 32, B), B, 0, stream>>>(acc2, ei, npair, Wlin, blin, out);
}